// GCN_30794915512600
// MI455X (gfx1250) — compile-verified
//
#include <hip/hip_runtime.h>
#include <hip/hip_bf16.h>
#include <math.h>

typedef __attribute__((ext_vector_type(2))) float v2f;
typedef __attribute__((ext_vector_type(8))) float v8f;

#define HDIM 128
#define CDIM 16

// ---------------------------------------------------------------- degree
__global__ void k_deg_init(int* __restrict__ deg, int n) {
    int i = blockIdx.x * blockDim.x + threadIdx.x;
    if (i < n) deg[i] = 1;                       // self-loop
}

__global__ void k_deg_count(const int* __restrict__ dst, int E, int* __restrict__ deg) {
    int e = blockIdx.x * blockDim.x + threadIdx.x;
    if (e < E) atomicAdd(&deg[dst[e]], 1);
}

__global__ void k_dinv(int* __restrict__ deg_in, float* __restrict__ dinv, int n) {
    int i = blockIdx.x * blockDim.x + threadIdx.x;
    if (i < n) {
        int d = deg_in[i];                        // read int
        dinv[i] = rsqrtf((float)d);               // write float (in place, per-thread)
    }
}

// ---------------------------------------------------------------- GEMM: H = X @ W  (X: Nx128, W: 128x128 row-major)
// One wave32 computes a 16-row x 128-col strip using V_WMMA_F32_16X16X4_F32.
// A 16x4 f32 layout: lanes 0-15 hold M=lane, VGPR0=K0,VGPR1=K1; lanes 16-31 hold K2,K3.
// B 4x16 f32 layout: VGPR v holds row K=v (lanes 0-15) / K=v+2 (lanes 16-31), col = lane&15.
// C 16x16 f32 layout: VGPR v holds row M=v (lanes 0-15) / M=v+8 (lanes 16-31), col = lane&15.
__global__ __launch_bounds__(256)
void k_gemm128(const float* __restrict__ X, const float* __restrict__ W,
               float* __restrict__ Hout, int nRows, int nTiles) {
    __shared__ float ldsW[HDIM * HDIM];          // 64 KB of the 320 KB WGP LDS
    for (int t = threadIdx.x; t < HDIM * HDIM / 4; t += 256)
        ((float4*)ldsW)[t] = ((const float4*)W)[t];
    __syncthreads();

    const int wave = threadIdx.x >> 5;
    const int lane = threadIdx.x & 31;
    const int tile = blockIdx.x * 8 + wave;
    if (tile >= nTiles) return;                  // whole wave exits: EXEC stays all-1s for WMMA

    int arow = tile * 16 + (lane & 15);
    if (arow >= nRows) arow = nRows - 1;         // clamp (N is a multiple of 16 here anyway)
    const int kh = (lane >> 4) * 2;              // 0 for lanes 0-15, 2 for lanes 16-31
    const int col0 = lane & 15;
    const float* xrow = X + (size_t)arow * HDIM;

    v8f acc[8] = {};
    for (int k = 0; k < HDIM; k += 4) {
        v2f a;
        a.x = xrow[k + kh];
        a.y = xrow[k + kh + 1];
#pragma unroll
        for (int t = 0; t < 8; ++t) {
            const int col = t * 16 + col0;
            v2f b;
            b.x = ldsW[(k + kh)     * HDIM + col];
            b.y = ldsW[(k + kh + 1) * HDIM + col];
            acc[t] = __builtin_amdgcn_wmma_f32_16x16x4_f32(
                false, a, false, b, (short)0, acc[t], false, false);
        }
    }

    const int rbase = tile * 16 + (lane >> 4) * 8;
#pragma unroll
    for (int v = 0; v < 8; ++v) {
        const int row = rbase + v;
        if (row < nRows) {
#pragma unroll
            for (int t = 0; t < 8; ++t)
                Hout[(size_t)row * HDIM + t * 16 + col0] = acc[t][v];
        }
    }
}

// ---------------------------------------------------------------- out = b + dinv^2 * h   (bias + self-loop term)
__global__ void k_init_out(const float* __restrict__ h, const float* __restrict__ bias,
                           const float* __restrict__ dinv, float* __restrict__ out, int n) {
    int idx = blockIdx.x * blockDim.x + threadIdx.x;   // over n * (HDIM/4) float4 chunks
    if (idx >= n * (HDIM / 4)) return;
    const int i = idx >> 5;          // node
    const int j = idx & 31;          // float4 chunk within row
    const float di = dinv[i];
    const float s = di * di;
    const float4 hv = ((const float4*)h)[idx];
    const float4 bv = ((const float4*)bias)[j];
    float4 o;
    o.x = bv.x + s * hv.x;
    o.y = bv.y + s * hv.y;
    o.z = bv.z + s * hv.z;
    o.w = bv.w + s * hv.w;
    ((float4*)out)[idx] = o;
}

// ---------------------------------------------------------------- edge scatter: out[dst] += norm * h[src]
// One wave32 per edge; each lane handles a float4 chunk of the 128-wide feature row.
__global__ __launch_bounds__(256)
void k_scatter(const int* __restrict__ src, const int* __restrict__ dst,
               const float* __restrict__ dinv, const float* __restrict__ h,
               float* __restrict__ out, int E) {
    const int t = blockIdx.x * blockDim.x + threadIdx.x;
    if (t >= E * 32) return;
    const int e = t >> 5;
    const int c = t & 31;
    const int s = src[e];
    const int d = dst[e];
    const float nrm = dinv[s] * dinv[d];
    const float4 hv = *(const float4*)(h + (size_t)s * HDIM + c * 4);
    float* op = out + (size_t)d * HDIM + c * 4;
    unsafeAtomicAdd(op + 0, nrm * hv.x);        // global_atomic_add_f32, no return
    unsafeAtomicAdd(op + 1, nrm * hv.y);
    unsafeAtomicAdd(op + 2, nrm * hv.z);
    unsafeAtomicAdd(op + 3, nrm * hv.w);
}

// ---------------------------------------------------------------- ELU in place
__global__ void k_elu(float* __restrict__ x, int n4) {
    int idx = blockIdx.x * blockDim.x + threadIdx.x;
    if (idx >= n4) return;
    float4 v = ((float4*)x)[idx];
    v.x = v.x > 0.0f ? v.x : expf(v.x) - 1.0f;
    v.y = v.y > 0.0f ? v.y : expf(v.y) - 1.0f;
    v.z = v.z > 0.0f ? v.z : expf(v.z) - 1.0f;
    v.w = v.w > 0.0f ? v.w : expf(v.w) - 1.0f;
    ((float4*)x)[idx] = v;
}

// ---------------------------------------------------------------- final FC + log_softmax
// 16 lanes (half a wave32) per node, one class per lane.
__global__ __launch_bounds__(256)
void k_final(const float* __restrict__ h, const float* __restrict__ Wfc,
             const float* __restrict__ bfc, float* __restrict__ out, int n) {
    __shared__ float lw[HDIM * CDIM];            // 8 KB
    for (int t = threadIdx.x; t < HDIM * CDIM / 4; t += 256)
        ((float4*)lw)[t] = ((const float4*)Wfc)[t];
    __syncthreads();

    const int half = threadIdx.x >> 4;           // 0..15
    const int c = threadIdx.x & 15;              // class
    const int node = blockIdx.x * 16 + half;
    if (node >= n) return;

    const float* hr = h + (size_t)node * HDIM;
    float logit = bfc[c];
    for (int k = 0; k < HDIM; ++k)
        logit += hr[k] * lw[k * CDIM + c];

    float m = logit;
    for (int off = 8; off >= 1; off >>= 1)
        m = fmaxf(m, __shfl_xor(m, off, 16));
    float ex = expf(logit - m);
    float ssum = ex;
    for (int off = 8; off >= 1; off >>= 1)
        ssum += __shfl_xor(ssum, off, 16);
    out[(size_t)node * CDIM + c] = logit - m - logf(ssum);
}

// ---------------------------------------------------------------- launch
extern "C" void kernel_launch(void* const* d_in, const int* in_sizes, int n_in,
                              void* d_out, int out_size, void* d_ws, size_t ws_size,
                              hipStream_t stream) {
    const float* x   = (const float*)d_in[0];
    const int*   ei  = (const int*)d_in[1];
    const float* W1  = (const float*)d_in[2];
    const float* b1  = (const float*)d_in[3];
    const float* W2  = (const float*)d_in[4];
    const float* b2  = (const float*)d_in[5];
    const float* W3  = (const float*)d_in[6];
    const float* b3  = (const float*)d_in[7];
    const float* Wfc = (const float*)d_in[8];
    const float* bfc = (const float*)d_in[9];

    const int N = in_sizes[0] / HDIM;
    const int E = in_sizes[1] / 2;
    const int* src = ei;
    const int* dst = ei + E;

    // workspace layout: dinv[N] | bufA[N*128] | bufB[N*128]
    char* ws = (char*)d_ws;
    size_t off = (((size_t)N * 4) + 255) & ~(size_t)255;
    float* dinv = (float*)ws;
    float* bufA = (float*)(ws + off);
    float* bufB = bufA + (size_t)N * HDIM;

    const int nTiles = (N + 15) / 16;
    const int n4 = N * (HDIM / 4);

    // degrees -> dinv
    k_deg_init<<<(N + 255) / 256, 256, 0, stream>>>((int*)dinv, N);
    k_deg_count<<<(E + 255) / 256, 256, 0, stream>>>(dst, E, (int*)dinv);
    k_dinv<<<(N + 255) / 256, 256, 0, stream>>>((int*)dinv, dinv, N);

    const float* layer_in[3] = { x, bufB, bufB };
    const float* Ws[3] = { W1, W2, W3 };
    const float* bs[3] = { b1, b2, b3 };
    for (int l = 0; l < 3; ++l) {
        k_gemm128<<<(nTiles + 7) / 8, 256, 0, stream>>>(layer_in[l], Ws[l], bufA, N, nTiles);
        k_init_out<<<(n4 + 255) / 256, 256, 0, stream>>>(bufA, bs[l], dinv, bufB, N);
        k_scatter<<<(E * 32 + 255) / 256, 256, 0, stream>>>(src, dst, dinv, bufA, bufB, E);
        k_elu<<<(n4 + 255) / 256, 256, 0, stream>>>(bufB, n4);
    }

    k_final<<<(N + 15) / 16, 256, 0, stream>>>(bufB, Wfc, bfc, (float*)d_out, N);
}